// RegionAttention_19653770347029
// MI455X (gfx1250) — compile-verified
//
#include <hip/hip_runtime.h>
#include <hip/hip_bf16.h>

typedef __attribute__((ext_vector_type(2))) float v2f;
typedef __attribute__((ext_vector_type(8))) float v8f;

#define Bb 2
#define Cc 48
#define C3 144
#define Dd 32
#define Hh 64
#define Ww 64
#define LL (Dd*Hh*Ww)        /* 131072 */
#define NT_PER_B (LL/16)     /* 8192 n-tiles of width 16 per batch */
#define CHUNK 2048           /* L-chunk per wave in attention reduction */

// ---------------------------------------------------------------------------
// K1: qkv = W1(144x48) @ x(48,L) + b1, via fp32 WMMA 16x16x4.
// One wave = one 16-wide column tile; x-tile kept in regs, reused for 9 m-tiles.
// ---------------------------------------------------------------------------
__global__ void __launch_bounds__(256) k_qkv_pointwise(
    const float* __restrict__ x, const float* __restrict__ w1,
    const float* __restrict__ b1, float* __restrict__ out)
{
  __shared__ float lds_w[C3*Cc];
  __shared__ float lds_b[C3];
  for (int i = threadIdx.x; i < C3*Cc; i += 256) lds_w[i] = w1[i];
  for (int i = threadIdx.x; i < C3;    i += 256) lds_b[i] = b1[i];
  __syncthreads();

  const int wid  = threadIdx.x >> 5;
  const int lane = threadIdx.x & 31;
  const int col  = lane & 15;        // A row / B col / D col
  const int hi   = lane >> 4;        // k-half select
  const long nt  = (long)blockIdx.x * 8 + wid;
  const int  b   = (int)(nt / NT_PER_B);
  const long l0  = (nt % NT_PER_B) * 16;
  const float* xb = x + (long)b * Cc * LL;

  v2f breg[12];
  #pragma unroll
  for (int ks = 0; ks < 12; ++ks) {
    const int kb = ks*4 + 2*hi;
    v2f t;
    t.x = xb[(long)(kb+0)*LL + l0 + col];
    t.y = xb[(long)(kb+1)*LL + l0 + col];
    breg[ks] = t;
  }

  float* ob = out + (long)b * C3 * LL + l0 + col;
  for (int mt = 0; mt < 9; ++mt) {
    v8f accA = {}, accB = {};
    #pragma unroll
    for (int ks = 0; ks < 6; ++ks) {
      v2f a0 = *(const v2f*)&lds_w[(mt*16 + col)*Cc + (2*ks+0)*4 + 2*hi];
      v2f a1 = *(const v2f*)&lds_w[(mt*16 + col)*Cc + (2*ks+1)*4 + 2*hi];
      accA = __builtin_amdgcn_wmma_f32_16x16x4_f32(false, a0, false, breg[2*ks+0],
                                                   (short)0, accA, false, false);
      accB = __builtin_amdgcn_wmma_f32_16x16x4_f32(false, a1, false, breg[2*ks+1],
                                                   (short)0, accB, false, false);
    }
    v8f acc = accA + accB;
    #pragma unroll
    for (int r = 0; r < 8; ++r) {
      const int mrow = mt*16 + r + 8*hi;
      ob[(long)mrow * LL] = acc[r] + lds_b[mrow];
    }
  }
}

// ---------------------------------------------------------------------------
// K2: depthwise 3x3x3 SAME conv, one (b,ch,d) slab per block, 3 halo planes
// (66x66) staged in LDS (52 KB), 27 VALU FMAs per output.
// ---------------------------------------------------------------------------
__global__ void __launch_bounds__(256) k_depthwise(
    const float* __restrict__ qkv1, const float* __restrict__ wdw,
    const float* __restrict__ bdw, float* __restrict__ out)
{
  __shared__ float pl[3][66*66];
  const int d  = blockIdx.x % Dd;
  const int bc = blockIdx.x / Dd;        // b*144 + ch
  const int ch = bc % C3;
  const float* src = qkv1 + (long)bc * LL;

  for (int pd = 0; pd < 3; ++pd) {
    const int dd = d - 1 + pd;
    const bool dok = (unsigned)dd < (unsigned)Dd;
    const float* sp = src + (long)dd * (Hh*Ww);
    for (int idx = threadIdx.x; idx < 66*66; idx += 256) {
      const int hy = idx / 66, wx = idx % 66;
      const int h = hy - 1, w = wx - 1;
      float v = 0.f;
      if (dok && (unsigned)h < (unsigned)Hh && (unsigned)w < (unsigned)Ww)
        v = sp[h*Ww + w];
      pl[pd][idx] = v;
    }
  }
  float wgt[27];
  #pragma unroll
  for (int t = 0; t < 27; ++t) wgt[t] = wdw[ch*27 + t];
  const float bias = bdw[ch];
  __syncthreads();

  float* dst = out + (long)bc * LL + (long)d * (Hh*Ww);
  for (int p = threadIdx.x; p < Hh*Ww; p += 256) {
    const int h = p >> 6, w = p & 63;
    float s = bias;
    #pragma unroll
    for (int kd = 0; kd < 3; ++kd)
      #pragma unroll
      for (int kh = 0; kh < 3; ++kh)
        #pragma unroll
        for (int kw = 0; kw < 3; ++kw)
          s += wgt[kd*9+kh*3+kw] * pl[kd][(h+kh)*66 + (w+kw)];
    dst[p] = s;
  }
}

__global__ void k_zero(float* __restrict__ p, int n) {
  const int i = blockIdx.x*256 + threadIdx.x;
  if (i < n) p[i] = 0.f;
}

// ---------------------------------------------------------------------------
// K3: S[b,pair,16,16] += q_pack(16,chunk) @ k_pack(chunk,16) via WMMA f32,
// contraction over L. Two heads packed per tile (channels contiguous), so all
// loads are useful. Two accumulators break the WMMA dependence chain.
// ---------------------------------------------------------------------------
__global__ void __launch_bounds__(256) k_attn_partial(
    const float* __restrict__ qkv, float* __restrict__ S)
{
  const int wid  = threadIdx.x >> 5, lane = threadIdx.x & 31;
  const int col  = lane & 15, hi = lane >> 4;
  const int wgl  = blockIdx.x * 8 + wid;           // 0..383
  const int nchunk = LL / CHUNK;                    // 64
  const int b    = wgl / (3*nchunk);
  const int rem  = wgl % (3*nchunk);
  const int pair = rem / nchunk;
  const long l0  = (long)(rem % nchunk) * CHUNK;

  const float* qrow = qkv + ((long)b*C3 +      pair*16 + col) * LL + l0 + 2*hi;
  const float* krow = qkv + ((long)b*C3 + 48 + pair*16 + col) * LL + l0 + 2*hi;

  v8f acc0 = {}, acc1 = {};
  for (int lo = 0; lo < CHUNK; lo += 64) {
    __builtin_prefetch(qrow + lo + 512, 0, 3);
    __builtin_prefetch(krow + lo + 512, 0, 3);
    #pragma unroll
    for (int l = 0; l < 64; l += 8) {
      v2f a0 = *(const v2f*)(qrow + lo + l);
      v2f b0 = *(const v2f*)(krow + lo + l);
      v2f a1 = *(const v2f*)(qrow + lo + l + 4);
      v2f b1 = *(const v2f*)(krow + lo + l + 4);
      acc0 = __builtin_amdgcn_wmma_f32_16x16x4_f32(false, a0, false, b0,
                                                   (short)0, acc0, false, false);
      acc1 = __builtin_amdgcn_wmma_f32_16x16x4_f32(false, a1, false, b1,
                                                   (short)0, acc1, false, false);
    }
  }
  v8f acc = acc0 + acc1;
  float* sp = S + ((long)b*3 + pair) * 256 + col;
  #pragma unroll
  for (int r = 0; r < 8; ++r)
    unsafeAtomicAdd(&sp[(r + 8*hi) * 16], acc[r]);   // global_atomic_add_f32
}

// ---------------------------------------------------------------------------
// K4 (tiny): scale + mask + softmax over j, then M_b = Wproj @ BlockDiag(attn).
// ---------------------------------------------------------------------------
__global__ void __launch_bounds__(256) k_finalize(
    const float* __restrict__ S, const float* __restrict__ mask,
    const float* __restrict__ wproj, float* __restrict__ M)
{
  __shared__ float attn[6*64];
  __shared__ float attn2[6*64];
  const int b = blockIdx.x;

  for (int t = threadIdx.x; t < 384; t += 256) {
    const int h = t / 64, i = (t / 8) % 8, j = t % 8;
    const int pair = h >> 1, half = h & 1;
    const float v = S[((long)b*3 + pair)*256 + (half*8 + i)*16 + (half*8 + j)];
    attn[t] = v * 0.40824829046386307f + mask[b*64 + i*8 + j];  // 6^-0.5
  }
  __syncthreads();
  if (threadIdx.x < 48) {
    const int base = threadIdx.x * 8;
    float mx = attn[base];
    #pragma unroll
    for (int j = 1; j < 8; ++j) mx = fmaxf(mx, attn[base+j]);
    float e[8]; float sm = 0.f;
    #pragma unroll
    for (int j = 0; j < 8; ++j) { e[j] = __expf(attn[base+j] - mx); sm += e[j]; }
    const float inv = 1.f / sm;
    #pragma unroll
    for (int j = 0; j < 8; ++j) attn2[base+j] = e[j] * inv;
  }
  __syncthreads();
  for (int idx = threadIdx.x; idx < 48*48; idx += 256) {
    const int co = idx / 48, cc = idx % 48;
    const int h = cc >> 3, j = cc & 7;
    float s = 0.f;
    #pragma unroll
    for (int i = 0; i < 8; ++i)
      s += wproj[co*48 + h*8 + i] * attn2[h*64 + i*8 + j];
    M[(long)b*2304 + idx] = s;
  }
}

// ---------------------------------------------------------------------------
// K5: out = M_b(48x48) @ v_dw(48,L) + b_proj, via fp32 WMMA (3 m-tiles).
// ---------------------------------------------------------------------------
__global__ void __launch_bounds__(256) k_out_proj(
    const float* __restrict__ qkv, const float* __restrict__ M,
    const float* __restrict__ bproj, float* __restrict__ out)
{
  __shared__ float lds_m[48*48];
  __shared__ float lds_b[48];
  const int wid  = threadIdx.x >> 5;
  const long nt  = (long)blockIdx.x * 8 + wid;
  const int  b   = (int)(nt / NT_PER_B);   // uniform across block (8192 % 8 == 0)
  for (int i = threadIdx.x; i < 48*48; i += 256) lds_m[i] = M[(long)b*2304 + i];
  if (threadIdx.x < 48) lds_b[threadIdx.x] = bproj[threadIdx.x];
  __syncthreads();

  const int lane = threadIdx.x & 31;
  const int col  = lane & 15, hi = lane >> 4;
  const long l0  = (nt % NT_PER_B) * 16;
  const float* vb = qkv + ((long)b*C3 + 96) * LL + l0 + col;

  v2f breg[12];
  #pragma unroll
  for (int ks = 0; ks < 12; ++ks) {
    const int kb = ks*4 + 2*hi;
    v2f t;
    t.x = vb[(long)(kb+0)*LL];
    t.y = vb[(long)(kb+1)*LL];
    breg[ks] = t;
  }

  float* ob = out + (long)b * Cc * LL + l0 + col;
  for (int mt = 0; mt < 3; ++mt) {
    v8f accA = {}, accB = {};
    #pragma unroll
    for (int ks = 0; ks < 6; ++ks) {
      v2f a0 = *(const v2f*)&lds_m[(mt*16 + col)*48 + (2*ks+0)*4 + 2*hi];
      v2f a1 = *(const v2f*)&lds_m[(mt*16 + col)*48 + (2*ks+1)*4 + 2*hi];
      accA = __builtin_amdgcn_wmma_f32_16x16x4_f32(false, a0, false, breg[2*ks+0],
                                                   (short)0, accA, false, false);
      accB = __builtin_amdgcn_wmma_f32_16x16x4_f32(false, a1, false, breg[2*ks+1],
                                                   (short)0, accB, false, false);
    }
    v8f acc = accA + accB;
    #pragma unroll
    for (int r = 0; r < 8; ++r) {
      const int mrow = mt*16 + r + 8*hi;
      ob[(long)mrow * LL] = acc[r] + lds_b[mrow];
    }
  }
}

// ---------------------------------------------------------------------------
extern "C" void kernel_launch(void* const* d_in, const int* in_sizes, int n_in,
                              void* d_out, int out_size, void* d_ws, size_t ws_size,
                              hipStream_t stream) {
  const float* x      = (const float*)d_in[0];
  const float* mask   = (const float*)d_in[1];
  const float* w_qkv1 = (const float*)d_in[2];
  const float* b_qkv1 = (const float*)d_in[3];
  const float* w_dw   = (const float*)d_in[4];
  const float* b_dw   = (const float*)d_in[5];
  const float* w_proj = (const float*)d_in[6];
  const float* b_proj = (const float*)d_in[7];
  float* out = (float*)d_out;

  float* qkv1  = (float*)d_ws;                       // B*144*L
  float* qkvdw = qkv1  + (size_t)Bb*C3*LL;           // B*144*L
  float* S     = qkvdw + (size_t)Bb*C3*LL;           // B*3*256
  float* M     = S     + (size_t)Bb*3*256;           // B*48*48

  k_zero<<<(Bb*3*256 + 255)/256, 256, 0, stream>>>(S, Bb*3*256);
  k_qkv_pointwise<<<(Bb*NT_PER_B)/8, 256, 0, stream>>>(x, w_qkv1, b_qkv1, qkv1);
  k_depthwise<<<Bb*C3*Dd, 256, 0, stream>>>(qkv1, w_dw, b_dw, qkvdw);
  k_attn_partial<<<(Bb*3*(LL/CHUNK))/8, 256, 0, stream>>>(qkvdw, S);
  k_finalize<<<Bb, 256, 0, stream>>>(S, mask, w_proj, M);
  k_out_proj<<<(Bb*NT_PER_B)/8, 256, 0, stream>>>(qkvdw, M, b_proj, out);
}